// PointConv_1692217115333
// MI455X (gfx1250) — compile-verified
//
#include <hip/hip_runtime.h>

// ---------------------------------------------------------------------------
// PointConv for MI455X (gfx1250, wave32)
//   B=8, N=4096, D=61, K=16 neighbors, C=64 (=3+61), W=16, OC=64, KK=1024
// Kernel 1: fused KNN (distance + top-16 select, batch coords in LDS)
// Kernel 2: fused gather + WeightNet MLP + agg (WMMA) + final linear (WMMA)
//   - k-major LDS layouts so every WMMA operand fetch is one ds_load_b64
//   - padded LDS rows to kill bank conflicts
//   - dual accumulators in the K=1024 GEMM for WMMA ILP
// ---------------------------------------------------------------------------

#define NPTS   4096
#define BATCH  8
#define DFEAT  61
#define CCH    64      // 3 + 61
#define KNN_K  16
#define WNET   16
#define OUTC   64
#define KKDIM  1024    // CCH * WNET

#define KPAD   18      // padded k-row (bank-conflict-free, keeps 8B align)
#define AGGPAD 1026    // padded AGG row

typedef float v2f __attribute__((ext_vector_type(2)));
typedef float v8f __attribute__((ext_vector_type(8)));

__device__ __forceinline__ float relu_f(float x) { return fmaxf(x, 0.0f); }

// ---------------------------------------------------------------------------
// Kernel 1: KNN. One query point per thread; whole batch's coords in LDS.
// ---------------------------------------------------------------------------
__global__ __launch_bounds__(256) void knn_kernel(
    const float* __restrict__ xyz,   // [B,3,N]
    int* __restrict__ knn_idx)       // [B,N,16]
{
    __shared__ float sx[NPTS];
    __shared__ float sy[NPTS];
    __shared__ float sz[NPTS];

    const int b     = blockIdx.x >> 4;          // 16 blocks per batch
    const int qbase = (blockIdx.x & 15) * 256;
    const float* xb = xyz + (size_t)b * 3 * NPTS;

    for (int i = threadIdx.x; i < NPTS; i += 256) {
        sx[i] = xb[i];
        sy[i] = xb[NPTS + i];
        sz[i] = xb[2 * NPTS + i];
    }
    __syncthreads();

    const int q = qbase + threadIdx.x;
    const float qx = sx[q], qy = sy[q], qz = sz[q];

    float bd[KNN_K];
    int   bi[KNN_K];
#pragma unroll
    for (int t = 0; t < KNN_K; ++t) { bd[t] = 3.4e38f; bi[t] = 0; }
    float dmax = 3.4e38f;
    int   imax = 0;

    for (int m = 0; m < NPTS; ++m) {
        const float dx = sx[m] - qx;
        const float dy = sy[m] - qy;
        const float dz = sz[m] - qz;
        const float d  = fmaf(dx, dx, fmaf(dy, dy, dz * dz));
        if (d < dmax) {
            // replace current-max slot (per-slot select keeps arrays in VGPRs)
#pragma unroll
            for (int t = 0; t < KNN_K; ++t) {
                if (t == imax) { bd[t] = d; bi[t] = m; }
            }
            float nm = -1.0f; int ni = 0;
#pragma unroll
            for (int t = 0; t < KNN_K; ++t) {
                if (bd[t] > nm) { nm = bd[t]; ni = t; }
            }
            dmax = nm; imax = ni;
        }
    }

    int* o = knn_idx + ((size_t)b * NPTS + q) * KNN_K;
#pragma unroll
    for (int t = 0; t < KNN_K; ++t) o[t] = bi[t];
}

// ---------------------------------------------------------------------------
// Kernel 2: fused per-16-point-tile pipeline. Block = 128 threads (4 waves).
// LDS (floats):
//   sNPT [16][64][KPAD]   k-major new_points  (p,c,k)   18432
//   sWT  [16][16][KPAD]   k-major weightnet   (p,w,k)    4608
//   sAGG [16][AGGPAD]     flattened agg       (p, c*16+w) 16416
// total ~154 KB dynamic LDS.
// ---------------------------------------------------------------------------
__global__ __launch_bounds__(128) void pointconv_fused(
    const float* __restrict__ xyz,     // [B,3,N]
    const float* __restrict__ points,  // [B,D,N]
    const int*   __restrict__ knn_idx, // [B,N,16]
    const float* __restrict__ w0, const float* __restrict__ b0,  // [8,3],[8]
    const float* __restrict__ w1, const float* __restrict__ b1,  // [8,8],[8]
    const float* __restrict__ w2, const float* __restrict__ b2,  // [16,8],[16]
    const float* __restrict__ lin_w,   // [64,1024]
    const float* __restrict__ lin_b,   // [64]
    float* __restrict__ out)           // [B,64,N]
{
    extern __shared__ float smem[];
    float* sNPT = smem;                         // 16*64*KPAD
    float* sWT  = sNPT + 16 * CCH * KPAD;       // 16*16*KPAD
    float* sAGG = sWT  + 16 * WNET * KPAD;      // 16*AGGPAD

    const int tid  = threadIdx.x;        // 0..127
    const int wave = tid >> 5;           // 0..3
    const int lane = tid & 31;

    const int b  = blockIdx.x >> 8;          // 256 tiles per batch
    const int n0 = (blockIdx.x & 255) * 16;  // first point of tile

    const float* xb = xyz    + (size_t)b * 3 * NPTS;
    const float* pb = points + (size_t)b * DFEAT * NPTS;

    // ---------------- Phase A: gather + WeightNet (2 (p,k) pairs per thread)
    for (int pair = tid; pair < 256; pair += 128) {
        const int p = pair >> 4;
        const int k = pair & 15;
        const int n = n0 + p;
        const int j = knn_idx[((size_t)b * NPTS + n) * KNN_K + k];

        const float rx = xb[j]            - xb[n];
        const float ry = xb[NPTS + j]     - xb[NPTS + n];
        const float rz = xb[2 * NPTS + j] - xb[2 * NPTS + n];

        // WeightNet MLP: 3 -> 8 -> 8 -> 16 (ReLU)
        float h0[8];
#pragma unroll
        for (int o = 0; o < 8; ++o) {
            h0[o] = relu_f(fmaf(w0[o * 3 + 0], rx,
                           fmaf(w0[o * 3 + 1], ry,
                           fmaf(w0[o * 3 + 2], rz, b0[o]))));
        }
        float h1[8];
#pragma unroll
        for (int o = 0; o < 8; ++o) {
            float a = b1[o];
#pragma unroll
            for (int i = 0; i < 8; ++i) a = fmaf(w1[o * 8 + i], h0[i], a);
            h1[o] = relu_f(a);
        }
        float* wcol = sWT + (p * WNET) * KPAD;      // [w][k], k-major
#pragma unroll
        for (int o = 0; o < WNET; ++o) {
            float a = b2[o];
#pragma unroll
            for (int i = 0; i < 8; ++i) a = fmaf(w2[o * 8 + i], h1[i], a);
            wcol[o * KPAD + k] = relu_f(a);
        }

        // new_points column k: [rel_xyz(3) | features(61)], k-major
        float* npc = sNPT + (p * CCH) * KPAD + k;
        npc[0 * KPAD] = rx; npc[1 * KPAD] = ry; npc[2 * KPAD] = rz;
#pragma unroll 4
        for (int c = 0; c < DFEAT; ++c) npc[(3 + c) * KPAD] = pb[c * NPTS + j];
    }
    __syncthreads();

    // ---------------- Phase B: AGG[p][c][w] = sum_k NP[p][k][c] * W[p][k][w]
    // WMMA f32 16x16x4: A = NP^T tile (16c x 4k), B = W tile (4k x 16w).
    // k-major layouts -> each operand fetch is one ds_load_b64.
    {
        const int lrow  = lane & 15;          // A row (c) / B col (w)
        const int khalf = (lane >> 4) * 2;    // K sub-offset per lane half
        for (int jj = 0; jj < 16; ++jj) {
            const int job = wave * 16 + jj;
            const int p   = job >> 2;
            const int ct  = job & 3;
            const float* aBase = sNPT + (p * CCH + ct * 16 + lrow) * KPAD + khalf;
            const float* bBase = sWT  + (p * WNET + lrow) * KPAD + khalf;

            v8f acc = {0.f, 0.f, 0.f, 0.f, 0.f, 0.f, 0.f, 0.f};
#pragma unroll
            for (int k0 = 0; k0 < 16; k0 += 4) {
                v2f a  = *(const v2f*)(aBase + k0);
                v2f bm = *(const v2f*)(bBase + k0);
                acc = __builtin_amdgcn_wmma_f32_16x16x4_f32(
                    false, a, false, bm, (short)0, acc, false, false);
            }
            // D layout: VGPR r, lanes 0-15 -> M=r, lanes 16-31 -> M=r+8
#pragma unroll
            for (int r = 0; r < 8; ++r) {
                const int cl = ct * 16 + r + ((lane >> 4) << 3);
                sAGG[p * AGGPAD + cl * WNET + lrow] = acc[r];
            }
        }
    }
    __syncthreads();

    // ---------------- Phase C: OUT(16p x 64oc) = AGG(16x1024) @ lin_w^T
    // Wave `wave` computes the 16-point x 16-channel tile; dual accumulators.
    {
        const int lrow  = lane & 15;          // A row (point) / B col (out ch)
        const int khalf = (lane >> 4) * 2;
        const int oc    = wave * 16 + lrow;
        const float* aRow = sAGG + lrow * AGGPAD + khalf;
        const float* lwp  = lin_w + (size_t)oc * KKDIM + khalf;

        v8f acc0 = {0.f, 0.f, 0.f, 0.f, 0.f, 0.f, 0.f, 0.f};
        v8f acc1 = {0.f, 0.f, 0.f, 0.f, 0.f, 0.f, 0.f, 0.f};
#pragma unroll 4
        for (int k0 = 0; k0 < KKDIM; k0 += 8) {
            v2f a0 = *(const v2f*)(aRow + k0);
            v2f b0v = *(const v2f*)(lwp + k0);
            acc0 = __builtin_amdgcn_wmma_f32_16x16x4_f32(
                false, a0, false, b0v, (short)0, acc0, false, false);
            v2f a1 = *(const v2f*)(aRow + k0 + 4);
            v2f b1v = *(const v2f*)(lwp + k0 + 4);
            acc1 = __builtin_amdgcn_wmma_f32_16x16x4_f32(
                false, a1, false, b1v, (short)0, acc1, false, false);
        }
        v8f acc = acc0 + acc1;

        const float bias = lin_b[oc];
#pragma unroll
        for (int r = 0; r < 8; ++r) {
            const int p = r + ((lane >> 4) << 3);
            float v = acc[r] + bias;
            v = (v > 0.0f) ? v : 0.1f * v;          // LeakyReLU(0.1)
            out[((size_t)b * OUTC + oc) * NPTS + (n0 + p)] = v;
        }
    }
}

// ---------------------------------------------------------------------------
extern "C" void kernel_launch(void* const* d_in, const int* in_sizes, int n_in,
                              void* d_out, int out_size, void* d_ws, size_t ws_size,
                              hipStream_t stream) {
    const float* xyz    = (const float*)d_in[0];
    const float* points = (const float*)d_in[1];
    const float* w0     = (const float*)d_in[2];
    const float* b0     = (const float*)d_in[3];
    const float* w1     = (const float*)d_in[4];
    const float* b1     = (const float*)d_in[5];
    const float* w2     = (const float*)d_in[6];
    const float* b2     = (const float*)d_in[7];
    const float* lin_w  = (const float*)d_in[8];
    const float* lin_b  = (const float*)d_in[9];
    float* out = (float*)d_out;

    int* knn_idx = (int*)d_ws;   // B*N*16 ints = 2 MB

    // Kernel 1: 8 batches x 16 chunks of 256 queries
    knn_kernel<<<BATCH * (NPTS / 256), 256, 0, stream>>>(xyz, knn_idx);

    // Kernel 2: one block per 16-point tile; ~154 KB dynamic LDS
    const size_t lds_bytes =
        (16 * CCH * KPAD + 16 * WNET * KPAD + 16 * AGGPAD) * sizeof(float);
    pointconv_fused<<<BATCH * (NPTS / 16), 128, lds_bytes, stream>>>(
        xyz, points, knn_idx, w0, b0, w1, b1, w2, b2, lin_w, lin_b, out);
}